// GPRGNN_52261162057811
// MI455X (gfx1250) — compile-verified
//
#include <hip/hip_runtime.h>

// ---------------- problem constants (dims fixed by reference) ----------------
#define FIN   500
#define HID   64
#define NCLS  47
#define CPAD  48
#define KSTEP 10
#define KCH   100          // W1 rows staged per LDS chunk
#define NCHK  5            // 500 / 100
#define W1STR 160          // v2f-pair row stride (floats); 160 % 64 == 32 -> conflict-free
#define W2STR 96           // 96 % 64 == 32 -> conflict-free
#define HSTR  68           // h-tile row stride (floats); 68 % 64 == 4 -> conflict-free

typedef __attribute__((ext_vector_type(2))) float v2f;
typedef __attribute__((ext_vector_type(8))) float v8f;

// ---------------- degree / norm precompute ----------------
__global__ void gpr_init_deg(float* __restrict__ deg, int n) {
    int i = blockIdx.x * blockDim.x + threadIdx.x;
    if (i < n) deg[i] = 1.0f;                 // self loop weight
}

__global__ void gpr_count_deg(const int* __restrict__ col, float* __restrict__ deg, int e) {
    int i = blockIdx.x * blockDim.x + threadIdx.x;
    if (i < e) atomicAdd(&deg[col[i]], 1.0f);
}

__global__ void gpr_dis(const float* __restrict__ deg, float* __restrict__ dis,
                        float* __restrict__ dis2, int n) {
    int i = blockIdx.x * blockDim.x + threadIdx.x;
    if (i < n) {
        float d = deg[i];                     // >= 1 always
        dis[i]  = rsqrtf(d);
        dis2[i] = 1.0f / d;                   // dis^2 (self-loop norm)
    }
}

__global__ void gpr_norm(const int* __restrict__ row, const int* __restrict__ col,
                         const float* __restrict__ dis, float* __restrict__ nrm, int e) {
    int i = blockIdx.x * blockDim.x + threadIdx.x;
    if (i < e) nrm[i] = dis[row[i]] * dis[col[i]];
}

// ---------------- fused MLP: zp = pad48( relu(x@W1+b1) @ W2 + b2 ) ----------------
// 256 threads = 8 waves; each wave owns one 16-row tile per grid-stride iteration.
// LDS weights are stored K-pair interleaved: (k,n) -> (k>>1)*STR + 2n + (k&1),
// so a B fragment is a single aligned ds_load_b64 straight into the WMMA operand.
__global__ __launch_bounds__(256)
void gpr_mlp_wmma(const float* __restrict__ x,  const float* __restrict__ W1,
                  const float* __restrict__ b1, const float* __restrict__ W2,
                  const float* __restrict__ b2, float* __restrict__ zp, int n) {
    __shared__ float sBig[8704];              // union: W1 chunk (50*160=8000) / 8x h-tiles (8*1088)
    __shared__ float sW2[32 * W2STR];         // W2 interleaved 32*96 = 3072

    const int tid    = threadIdx.x;
    const int wid    = tid >> 5;
    const int lane   = tid & 31;
    const int rloc   = lane & 15;             // row-in-tile (GEMM1 A) / col (B, C)
    const int khalf  = lane >> 4;             // 0: K={0,1}, 1: K={2,3}

    // stage padded + interleaved W2 once
    for (int i = tid; i < HID * CPAD; i += 256) {
        int k = i / CPAD, c = i % CPAD;
        sW2[(k >> 1) * W2STR + 2 * c + (k & 1)] = (c < NCLS) ? W2[k * NCLS + c] : 0.0f;
    }

    const int numTiles = (n + 15) / 16;
    for (int base = blockIdx.x * 8; base < numTiles; base += gridDim.x * 8) {
        const int  tile   = base + wid;
        const bool active = tile < numTiles;  // wave-uniform

        v8f acc[4] = {};                      // 16x64 fp32 h tile
        int row = tile * 16 + rloc;
        if (row >= n) row = n - 1;

        // ---- GEMM1: x(16x500) @ W1(500x64), K chunked through LDS ----
        for (int kc = 0; kc < NCHK; ++kc) {
            __syncthreads();                  // prev chunk / prev-iter hs reads done
            for (int i = tid; i < KCH * HID; i += 256) {
                int k = i >> 6, c = i & 63;   // W1 row-major [k][c]
                sBig[(k >> 1) * W1STR + 2 * c + (k & 1)] = W1[kc * KCH * HID + i];
            }
            __syncthreads();
            if (active) {
                const float* xrow = x + (size_t)row * FIN + kc * KCH + 2 * khalf;
                const v2f* bmat = (const v2f*)sBig;
#pragma unroll
                for (int kk = 0; kk < KCH / 4; ++kk) {           // 25 its, fully unrolled
                    v2f a = ((const v2f*)xrow)[2 * kk];          // K = 4kk+2*khalf +{0,1}
                    const int pr = (2 * kk + khalf) * (W1STR / 2);
#pragma unroll
                    for (int nt = 0; nt < 4; ++nt) {
                        v2f b = bmat[pr + nt * 16 + rloc];       // single b64 LDS load
                        acc[nt] = __builtin_amdgcn_wmma_f32_16x16x4_f32(
                            false, a, false, b, (short)0, acc[nt], false, false);
                    }
                }
            }
        }

        // ---- bias + relu, stage h tile (stride-68 rows) in LDS ----
        __syncthreads();
        float* hsw = sBig + wid * (16 * HSTR);
        if (active) {
#pragma unroll
            for (int nt = 0; nt < 4; ++nt) {
                const float bb = b1[nt * 16 + rloc];
#pragma unroll
                for (int v = 0; v < 8; ++v) {
                    float val = acc[nt][v] + bb;
                    val = fmaxf(val, 0.0f);
                    const int M = v + 8 * khalf;       // C-layout row
                    hsw[M * HSTR + nt * 16 + rloc] = val;
                }
            }
        }
        __syncthreads();

        // ---- GEMM2: h(16x64) @ W2(64x48) ----
        if (active) {
            v8f acc2[3] = {};
            const v2f* amat = (const v2f*)hsw;         // wid*1088 even -> 8B aligned
            const v2f* bmat = (const v2f*)sW2;
#pragma unroll 4
            for (int kk = 0; kk < HID / 4; ++kk) {     // 16 its, unroll 4 -> no remainder
                v2f a2 = amat[rloc * (HSTR / 2) + 2 * kk + khalf];
                const int pr = (2 * kk + khalf) * (W2STR / 2);
#pragma unroll
                for (int nt = 0; nt < 3; ++nt) {
                    v2f b = bmat[pr + nt * 16 + rloc];
                    acc2[nt] = __builtin_amdgcn_wmma_f32_16x16x4_f32(
                        false, a2, false, b, (short)0, acc2[nt], false, false);
                }
            }
#pragma unroll
            for (int nt = 0; nt < 3; ++nt) {
                const int cb  = nt * 16 + rloc;
                const float bb = (cb < NCLS) ? b2[cb] : 0.0f;
#pragma unroll
                for (int v = 0; v < 8; ++v) {
                    const int r = tile * 16 + v + 8 * khalf;
                    if (r < n) {
                        float val = (cb < NCLS) ? (acc2[nt][v] + bb) : 0.0f; // zero pad col
                        zp[(size_t)r * CPAD + cb] = val;
                    }
                }
            }
        }
    }
}

// ---------------- log_softmax over 47 classes ----------------
__global__ void gpr_log_softmax(const float* __restrict__ zp, float* __restrict__ out, int n) {
    int i = blockIdx.x * blockDim.x + threadIdx.x;
    if (i >= n) return;
    const float* zr = zp + (size_t)i * CPAD;
    float vals[NCLS];
    float m = -3.4e38f;
#pragma unroll
    for (int c = 0; c < NCLS; ++c) { vals[c] = zr[c]; m = fmaxf(m, vals[c]); }
    float s = 0.0f;
#pragma unroll
    for (int c = 0; c < NCLS; ++c) s += __expf(vals[c] - m);
    const float ls = __logf(s);
    float* o = out + (size_t)i * NCLS;
#pragma unroll
    for (int c = 0; c < NCLS; ++c) o[c] = vals[c] - m - ls;
}

// ---------------- fused: hidden (+)= temp[k]*src ; dst = dis2*src (self loop) ----------------
__global__ void gpr_acc_selfloop(const float* __restrict__ src, float* __restrict__ dst,
                                 const float* __restrict__ dis2, const float* __restrict__ temp,
                                 float* __restrict__ hid, int k, int first, int n) {
    int idx = blockIdx.x * blockDim.x + threadIdx.x;
    if (idx >= n * CPAD) return;
    const int i = idx / CPAD, c = idx % CPAD;
    const float v = src[idx];
    dst[idx] = dis2[i] * v;                   // pad col stays 0 (src pad is 0)
    if (c < NCLS) {
        const float t = temp[k] * v;
        float* h = hid + (size_t)i * NCLS + c;
        if (first) *h = t; else *h += t;
    }
}

// ---------------- edge scatter: nxt[col] += norm*cur[row], rows padded to 48 ----------------
__global__ void gpr_scatter(const float* __restrict__ cur, float* __restrict__ nxt,
                            const int* __restrict__ row, const int* __restrict__ col,
                            const float* __restrict__ nrm, int e) {
    int ei = blockIdx.x * blockDim.x + threadIdx.x;
    if (ei >= e) return;
    const float w = nrm[ei];
    const float4* s = (const float4*)(cur + (size_t)row[ei] * CPAD);
    float* d = nxt + (size_t)col[ei] * CPAD;
#pragma unroll
    for (int j = 0; j < CPAD / 4; ++j) {
        float4 v = s[j];
        atomicAdd(d + 4 * j + 0, w * v.x);
        atomicAdd(d + 4 * j + 1, w * v.y);
        atomicAdd(d + 4 * j + 2, w * v.z);
        atomicAdd(d + 4 * j + 3, w * v.w);
    }
}

// ---------------- host ----------------
extern "C" void kernel_launch(void* const* d_in, const int* in_sizes, int n_in,
                              void* d_out, int out_size, void* d_ws, size_t ws_size,
                              hipStream_t stream) {
    const float* x    = (const float*)d_in[0];
    const int*   ei   = (const int*)  d_in[1];
    const float* W1   = (const float*)d_in[2];
    const float* b1   = (const float*)d_in[3];
    const float* W2   = (const float*)d_in[4];
    const float* b2   = (const float*)d_in[5];
    const float* temp = (const float*)d_in[6];

    const int N = in_sizes[0] / FIN;
    const int E = in_sizes[1] / 2;
    const int* row = ei;
    const int* col = ei + E;

    float* ws   = (float*)d_ws;
    const size_t N48 = (size_t)N * CPAD;
    float* zp   = ws;
    float* bufA = ws + N48;
    float* bufB = ws + 2 * N48;
    float* deg  = ws + 3 * N48;
    float* dis  = deg + N;
    float* dis2 = dis + N;
    float* nrm  = dis2 + N;

    float* out_ls = (float*)d_out;
    float* hid    = out_ls + (size_t)N * NCLS;

    const int TB = 256;
    const int gN = (N + TB - 1) / TB;
    const int gE = (E + TB - 1) / TB;

    // gcn_norm
    gpr_init_deg <<<gN, TB, 0, stream>>>(deg, N);
    gpr_count_deg<<<gE, TB, 0, stream>>>(col, deg, E);
    gpr_dis      <<<gN, TB, 0, stream>>>(deg, dis, dis2, N);
    gpr_norm     <<<gE, TB, 0, stream>>>(row, col, dis, nrm, E);

    // fused MLP (WMMA), then log_softmax half of the output
    const int numTiles = (N + 15) / 16;
    const int gG = (numTiles + 7) / 8;
    gpr_mlp_wmma <<<gG, TB, 0, stream>>>(x, W1, b1, W2, b2, zp, N);
    gpr_log_softmax<<<gN, TB, 0, stream>>>(zp, out_ls, N);

    // GPR propagation: hidden = sum_k temp[k] * A_hat^k z
    const int gP = ((N * CPAD) + TB - 1) / TB;
    gpr_acc_selfloop<<<gP, TB, 0, stream>>>(zp, bufA, dis2, temp, hid, 0, 1, N);
    const float* cur = zp;
    float* nx = bufA;
    for (int k = 1; k <= KSTEP; ++k) {
        gpr_scatter<<<gE, TB, 0, stream>>>(cur, nx, row, col, nrm, E);
        float* nn = (nx == bufA) ? bufB : bufA;   // never recycles zp
        gpr_acc_selfloop<<<gP, TB, 0, stream>>>(nx, nn, dis2, temp, hid, k, 0, N);
        cur = nx;
        nx = nn;
    }
}